// PointNet2SemSegDittoMsg_21895743275307
// MI455X (gfx1250) — compile-verified
//
#include <hip/hip_runtime.h>

// ---------------------------------------------------------------------------
// PointNet++ MSG seg forward for MI455X (gfx1250, wave32, WMMA).
// All dense layers use V_WMMA_F32_16X16X4_F32 (fp32 A/B, fp32 accum).
// Weights pre-packed (zero-padded, row-pair interleaved) -> inner WMMA loops
// are branch-free and software-pipelined: loads for step k+1 issue before
// the WMMAs of step k, so the per-iteration s_wait covers a full loop body.
// ---------------------------------------------------------------------------

typedef float v2f __attribute__((ext_vector_type(2)));
typedef float v8f __attribute__((ext_vector_type(8)));

__device__ __forceinline__ v8f wmma_f32(v2f a, v2f b, v8f c) {
  // 8 args: (neg_a, A, neg_b, B, c_mod, C, reuse_a, reuse_b)
  return __builtin_amdgcn_wmma_f32_16x16x4_f32(false, a, false, b, (short)0, c,
                                               false, false);
}

// ---------------------------------------------------------------------------
// 0) Weight packing: W (cin x cout) -> Wp (pad4(cin) x pad16(cout)),
//    zero-filled, rows interleaved in pairs so the B fragment (rows ak,ak+1
//    at column j; ak always even) is one contiguous b64:
//      Wp[(i>>1)*(2*coutP) + 2*j + (i&1)] = W[i][j]
// ---------------------------------------------------------------------------
__global__ void pack_weight_kernel(const float* __restrict__ W,
                                   float* __restrict__ Wp, int cin, int cout,
                                   int cinP, int coutP) {
  int t = blockIdx.x * blockDim.x + threadIdx.x;
  if (t >= cinP * coutP) return;
  int i = t / coutP;
  int j = t % coutP;
  float v = (i < cin && j < cout) ? W[(size_t)i * cout + j] : 0.0f;
  Wp[(size_t)(i >> 1) * (2 * coutP) + 2 * j + (i & 1)] = v;
}

// ---------------------------------------------------------------------------
// 1) Input transpose: xyz (B,6,N) -> l1_xyz (B,N,3), l1_points (B,N,6)
// ---------------------------------------------------------------------------
__global__ void prep_kernel(const float* __restrict__ xyz,
                            float* __restrict__ l1_xyz,
                            float* __restrict__ l1_pts, int B, int C, int N) {
  int t = blockIdx.x * blockDim.x + threadIdx.x;
  int total = B * C * N;
  if (t >= total) return;
  int n = t % N;
  int c = (t / N) % C;
  int b = t / (N * C);
  float v = xyz[t];
  l1_pts[((size_t)b * N + n) * C + c] = v;
  if (c < 3) l1_xyz[((size_t)b * N + n) * 3 + c] = v;
}

// ---------------------------------------------------------------------------
// 2) Farthest point sampling. One block per batch; distance array in LDS.
// ---------------------------------------------------------------------------
__global__ void fps_kernel(const float* __restrict__ xyz,
                           float* __restrict__ out_xyz, int N, int npoint) {
  extern __shared__ float smem[];
  float* sd = smem;
  float* rv = sd + N;
  int* ri = (int*)(rv + blockDim.x);
  int b = blockIdx.x;
  const float* px = xyz + (size_t)b * N * 3;
  for (int n = threadIdx.x; n < N; n += blockDim.x) sd[n] = 1e10f;
  __syncthreads();
  int cur = 0;
  for (int j = 0; j < npoint; ++j) {
    if (threadIdx.x < 3)
      out_xyz[((size_t)b * npoint + j) * 3 + threadIdx.x] =
          px[cur * 3 + threadIdx.x];
    float cx = px[cur * 3 + 0], cy = px[cur * 3 + 1], cz = px[cur * 3 + 2];
    float bv = -1.0f;
    int bi = 0;
    for (int n = threadIdx.x; n < N; n += blockDim.x) {
      float dx = px[n * 3 + 0] - cx;
      float dy = px[n * 3 + 1] - cy;
      float dz = px[n * 3 + 2] - cz;
      float d = dx * dx + dy * dy + dz * dz;
      float nd = fminf(sd[n], d);
      sd[n] = nd;
      if (nd > bv) { bv = nd; bi = n; }
    }
    rv[threadIdx.x] = bv;
    ri[threadIdx.x] = bi;
    __syncthreads();
    for (int off = blockDim.x >> 1; off > 0; off >>= 1) {
      if (threadIdx.x < off) {
        float v2 = rv[threadIdx.x + off];
        int i2 = ri[threadIdx.x + off];
        if (v2 > rv[threadIdx.x] ||
            (v2 == rv[threadIdx.x] && i2 < ri[threadIdx.x])) {
          rv[threadIdx.x] = v2;
          ri[threadIdx.x] = i2;
        }
      }
      __syncthreads();
    }
    cur = ri[0];
    __syncthreads();
  }
}

// ---------------------------------------------------------------------------
// 3) Ball query: first K indices (ascending) within radius, pad with first.
// ---------------------------------------------------------------------------
__global__ void ball_query_kernel(const float* __restrict__ xyz,
                                  const float* __restrict__ new_xyz,
                                  int* __restrict__ idx, int B, int N, int S,
                                  int K, float r2) {
  int t = blockIdx.x * blockDim.x + threadIdx.x;
  if (t >= B * S) return;
  int b = t / S;
  const float* px = xyz + (size_t)b * N * 3;
  float cx = new_xyz[t * 3 + 0], cy = new_xyz[t * 3 + 1],
        cz = new_xyz[t * 3 + 2];
  int* oi = idx + (size_t)t * K;
  int cnt = 0, first = 0;
  for (int n = 0; n < N && cnt < K; ++n) {
    float dx = px[n * 3 + 0] - cx;
    float dy = px[n * 3 + 1] - cy;
    float dz = px[n * 3 + 2] - cz;
    if (dx * dx + dy * dy + dz * dz <= r2) {
      if (cnt == 0) first = n;
      oi[cnt++] = n;
    }
  }
  for (; cnt < K; ++cnt) oi[cnt] = first;
}

// ---------------------------------------------------------------------------
// 4) Fused SA branch: gather group (points || centered xyz) -> 3 WMMA MLP
//    layers (relu((x@W+b)*s+bb)) in LDS ping-pong -> max over K -> out.
//    Block = one (batch, centroid); 4 waves, each wave owns 16x16 tiles.
// ---------------------------------------------------------------------------
struct SAParams {
  const float* Wp[3];  // packed weights
  const float* bp[3];
  const float* sp[3];
  const float* zp[3];
  int cin[3];
  int cout[3];
  int coP[3];  // pad16(cout)
};

#define SASTRIDE 264  // >= pad4(259), even, dodges bank conflicts

__global__ void sa_branch_kernel(const float* __restrict__ xyz,
                                 const float* __restrict__ points,
                                 const float* __restrict__ new_xyz,
                                 const int* __restrict__ idx, int N, int S,
                                 int K, int cp, SAParams p,
                                 float* __restrict__ outPts, int chOff,
                                 int Ctot) {
  extern __shared__ float smem[];
  float* bufA = smem;
  float* bufB = smem + 32 * SASTRIDE;
  int b = blockIdx.x / S;
  int s = blockIdx.x % S;
  const int tid = threadIdx.x;
  const int nth = blockDim.x;
  const int* gi = idx + ((size_t)b * S + s) * K;
  const float* cxyz = new_xyz + ((size_t)b * S + s) * 3;

  int cin0 = cp + 3;
  int cin0p = (cin0 + 3) & ~3;
  for (int t = tid; t < K * cin0p; t += nth) {
    int k = t / cin0p, c = t % cin0p;
    float v = 0.0f;
    if (c < cin0) {
      int pi = gi[k];
      if (c < cp)
        v = points[((size_t)b * N + pi) * cp + c];
      else
        v = xyz[((size_t)b * N + pi) * 3 + (c - cp)] - cxyz[c - cp];
    }
    bufA[k * SASTRIDE + c] = v;
  }
  __syncthreads();

  const int lane = tid & 31;
  const int wv = tid >> 5;
  const int nw = nth >> 5;
  const int half = lane >> 4;
  const int ml = lane & 15;
  float* cur = bufA;
  float* nxt = bufB;

  for (int l = 0; l < 3; ++l) {
    int cin = p.cin[l], cout = p.cout[l], coutP = p.coP[l];
    int cinP = (cin + 3) & ~3;
    int coutP4 = (cout + 3) & ~3;
    // zero next buffer's pad4 columns (next layer's A frags read them)
    for (int t = tid; t < K * (coutP4 - cout); t += nth) {
      int k = t / (coutP4 - cout);
      int c = cout + t % (coutP4 - cout);
      nxt[k * SASTRIDE + c] = 0.0f;
    }
    const float* bi = p.bp[l];
    const float* sc = p.sp[l];
    const float* sb = p.zp[l];
    int TR = K >> 4;
    int TC = coutP >> 4;
    for (int tile = wv; tile < TR * TC; tile += nw) {  // wave-uniform
      int rt = tile / TC, ct = tile % TC;
      int rowBase = rt * 16, colBase = ct * 16;
      const float* Wl = p.Wp[l] + 2 * (colBase + ml) + (size_t)half * 2 * coutP;
      const float* Al = cur + (rowBase + ml) * SASTRIDE + 2 * half;
      v8f acc = {0, 0, 0, 0, 0, 0, 0, 0};
      // software pipeline: fetch k+1 fragments before WMMA of step k
      v2f aC = *(const v2f*)(Al);
      v2f bC = *(const v2f*)(Wl);
      for (int kk = 4; kk < cinP; kk += 4) {
        v2f aN = *(const v2f*)(Al + kk);
        v2f bN = *(const v2f*)(Wl + (size_t)(kk >> 1) * 2 * coutP);
        acc = wmma_f32(aC, bC, acc);
        aC = aN;
        bC = bN;
      }
      acc = wmma_f32(aC, bC, acc);
      int j = colBase + ml;
      if (j < cout) {
        float bv = bi[j], sv = sc[j], zv = sb[j];
#pragma unroll
        for (int e = 0; e < 8; ++e) {  // C/D: element e -> row e+8*half
          int row = rowBase + e + 8 * half;
          float x = (acc[e] + bv) * sv + zv;
          nxt[row * SASTRIDE + j] = fmaxf(x, 0.0f);
        }
      }
    }
    __syncthreads();
    float* tmp = cur;
    cur = nxt;
    nxt = tmp;
  }
  int coutL = p.cout[2];
  for (int j = tid; j < coutL; j += nth) {
    float mx = cur[j];
    for (int k = 1; k < K; ++k) mx = fmaxf(mx, cur[k * SASTRIDE + j]);
    outPts[((size_t)b * S + s) * Ctot + chOff + j] = mx;
  }
}

// ---------------------------------------------------------------------------
// 5) 3-NN (squared distance) + normalized inverse-distance weights.
// ---------------------------------------------------------------------------
__global__ void knn3_kernel(const float* __restrict__ xyz1,
                            const float* __restrict__ xyz2, int B, int N1,
                            int N2, int* __restrict__ nnIdx,
                            float* __restrict__ nnW) {
  int t = blockIdx.x * blockDim.x + threadIdx.x;
  if (t >= B * N1) return;
  int b = t / N1;
  float cx = xyz1[t * 3 + 0], cy = xyz1[t * 3 + 1], cz = xyz1[t * 3 + 2];
  const float* p2 = xyz2 + (size_t)b * N2 * 3;
  float d0 = 3.4e38f, d1 = 3.4e38f, d2 = 3.4e38f;
  int i0 = 0, i1 = 0, i2 = 0;
  for (int n = 0; n < N2; ++n) {
    float dx = p2[n * 3 + 0] - cx;
    float dy = p2[n * 3 + 1] - cy;
    float dz = p2[n * 3 + 2] - cz;
    float d = dx * dx + dy * dy + dz * dz;
    if (d < d0) {
      d2 = d1; i2 = i1; d1 = d0; i1 = i0; d0 = d; i0 = n;
    } else if (d < d1) {
      d2 = d1; i2 = i1; d1 = d; i1 = n;
    } else if (d < d2) {
      d2 = d; i2 = n;
    }
  }
  float w0 = 1.0f / (d0 + 1e-8f);
  float w1 = 1.0f / (d1 + 1e-8f);
  float w2 = 1.0f / (d2 + 1e-8f);
  float ws = w0 + w1 + w2;
  nnW[t * 3 + 0] = w0 / ws;
  nnW[t * 3 + 1] = w1 / ws;
  nnW[t * 3 + 2] = w2 / ws;
  nnIdx[t * 3 + 0] = b * N2 + i0;
  nnIdx[t * 3 + 1] = b * N2 + i1;
  nnIdx[t * 3 + 2] = b * N2 + i2;
}

// ---------------------------------------------------------------------------
// 6) WMMA MLP layer: Y = relu((X@W + b) * s + bb). Block = 16 rows x 128
//    cols: 4 waves, each wave computes TWO 16x16 tiles from one A frag.
//    X==nullptr -> A tile staged as fused concat of P1 and 3-NN interp of
//    P2 (no materialized concat buffer). Inner loops software-pipelined.
// ---------------------------------------------------------------------------
__global__ void mlp_layer_kernel(
    const float* __restrict__ X, const float* __restrict__ P1, int c1,
    const float* __restrict__ P2, int c2, const int* __restrict__ nnIdx,
    const float* __restrict__ nnW, const float* __restrict__ Wp,
    const float* __restrict__ bi, const float* __restrict__ sc,
    const float* __restrict__ sb, float* __restrict__ Y, int R, int cin,
    int cout, int coutP) {
  extern __shared__ float smem[];
  int cinP = (cin + 3) & ~3;
  int lstr = cinP + 4;  // even
  int rowBase = blockIdx.x * 16;
  int tid = threadIdx.x, nth = blockDim.x;
  for (int t = tid; t < 16 * cinP; t += nth) {
    int rr = t / cinP, c = t % cinP;
    int r = rowBase + rr;
    float v = 0.0f;
    if (r < R && c < cin) {
      if (X) {
        v = X[(size_t)r * cin + c];
      } else if (c < c1) {
        v = P1[(size_t)r * c1 + c];
      } else {
        int cc = c - c1;
        int j0 = nnIdx[r * 3 + 0], j1 = nnIdx[r * 3 + 1],
            j2 = nnIdx[r * 3 + 2];
        v = nnW[r * 3 + 0] * P2[(size_t)j0 * c2 + cc] +
            nnW[r * 3 + 1] * P2[(size_t)j1 * c2 + cc] +
            nnW[r * 3 + 2] * P2[(size_t)j2 * c2 + cc];
      }
    }
    smem[rr * lstr + c] = v;
  }
  __syncthreads();
  int lane = tid & 31, wv = tid >> 5;
  int half = lane >> 4, ml = lane & 15;
  int colBase = (blockIdx.y * (nth >> 5) + wv) * 32;  // wave-uniform
  if (colBase >= coutP) return;  // whole wave exits together
  const float* Al = smem + ml * lstr + 2 * half;
  const float* W0 = Wp + 2 * (colBase + ml) + (size_t)half * 2 * coutP;
  bool two = (colBase + 16) < coutP;
  v8f acc0 = {0, 0, 0, 0, 0, 0, 0, 0};
  v8f acc1 = {0, 0, 0, 0, 0, 0, 0, 0};
  if (two) {
    v2f aC = *(const v2f*)(Al);
    v2f b0C = *(const v2f*)(W0);
    v2f b1C = *(const v2f*)(W0 + 32);
    for (int kk = 4; kk < cinP; kk += 4) {
      const float* Wr = W0 + (size_t)(kk >> 1) * 2 * coutP;
      v2f aN = *(const v2f*)(Al + kk);
      v2f b0N = *(const v2f*)(Wr);
      v2f b1N = *(const v2f*)(Wr + 32);
      acc0 = wmma_f32(aC, b0C, acc0);
      acc1 = wmma_f32(aC, b1C, acc1);
      aC = aN;
      b0C = b0N;
      b1C = b1N;
    }
    acc0 = wmma_f32(aC, b0C, acc0);
    acc1 = wmma_f32(aC, b1C, acc1);
  } else {
    v2f aC = *(const v2f*)(Al);
    v2f b0C = *(const v2f*)(W0);
    for (int kk = 4; kk < cinP; kk += 4) {
      v2f aN = *(const v2f*)(Al + kk);
      v2f b0N = *(const v2f*)(W0 + (size_t)(kk >> 1) * 2 * coutP);
      acc0 = wmma_f32(aC, b0C, acc0);
      aC = aN;
      b0C = b0N;
    }
    acc0 = wmma_f32(aC, b0C, acc0);
  }
#pragma unroll
  for (int tI = 0; tI < 2; ++tI) {
    if (tI == 1 && !two) break;
    int j = colBase + tI * 16 + ml;
    v8f acc = tI ? acc1 : acc0;
    if (j < cout) {
      float bv = bi[j], sv = sc[j], zv = sb[j];
#pragma unroll
      for (int e = 0; e < 8; ++e) {
        int row = rowBase + e + 8 * half;
        if (row < R) {
          float x = (acc[e] + bv) * sv + zv;
          Y[(size_t)row * cout + j] = fmaxf(x, 0.0f);
        }
      }
    }
  }
}

// ---------------------------------------------------------------------------
// 7) Head2: sigmoid(h @ W2 + b2), cout = 1.
// ---------------------------------------------------------------------------
__global__ void head2_kernel(const float* __restrict__ H,
                             const float* __restrict__ W2,
                             const float* __restrict__ b2,
                             float* __restrict__ out, int R, int C) {
  int t = blockIdx.x * blockDim.x + threadIdx.x;
  if (t >= R) return;
  float acc = 0.0f;
  for (int c = 0; c < C; ++c) acc += H[(size_t)t * C + c] * W2[c];
  acc += b2[0];
  out[t] = 1.0f / (1.0f + __expf(-acc));
}

// ---------------------------------------------------------------------------
// 8) Output 2: l3_points (B,128,512) -> transpose(0,2,1) -> (B,512,128)
// ---------------------------------------------------------------------------
__global__ void out2_kernel(const float* __restrict__ l3p,
                            float* __restrict__ out2, int B, int Np, int Cc) {
  int t = blockIdx.x * blockDim.x + threadIdx.x;
  if (t >= B * Np * Cc) return;
  int n = t % Np;
  int c = (t / Np) % Cc;
  int b = t / (Np * Cc);
  out2[t] = l3p[((size_t)b * Np + n) * Cc + c];
}

// ---------------------------------------------------------------------------
// Host orchestration
// ---------------------------------------------------------------------------
extern "C" void kernel_launch(void* const* d_in, const int* in_sizes, int n_in,
                              void* d_out, int out_size, void* d_ws,
                              size_t ws_size, hipStream_t stream) {
  (void)in_sizes; (void)n_in; (void)out_size; (void)ws_size;
  const int B = 8, N = 8192, C = 6;
  const float* xyz = (const float*)d_in[0];
  auto F = [&](int i) { return (const float*)d_in[i]; };

  // workspace layout (fp32 / int32)
  float* w = (float*)d_ws;
  size_t o = 0;
  auto alloc = [&](size_t n) {
    float* p = w + o;
    o += (n + 63) & ~(size_t)63;
    return p;
  };
  float* l1_xyz = alloc((size_t)B * N * 3);
  float* l1_pts = alloc((size_t)B * N * 6);
  float* l2_xyz = alloc((size_t)B * 512 * 3);
  float* l2_pts = alloc((size_t)B * 512 * 256);
  float* l3_xyz = alloc((size_t)B * 128 * 3);
  float* l3_pts = alloc((size_t)B * 128 * 512);
  int* idx2A = (int*)alloc((size_t)B * 512 * 16);
  int* idx2B = (int*)alloc((size_t)B * 512 * 32);
  int* idx3A = (int*)alloc((size_t)B * 128 * 16);
  int* idx3B = (int*)alloc((size_t)B * 128 * 32);
  int* nn3i = (int*)alloc((size_t)B * 512 * 3);
  float* nn3w = alloc((size_t)B * 512 * 3);
  float* fp3h1 = alloc((size_t)B * 512 * 256);
  float* fp3h2 = alloc((size_t)B * 512 * 256);
  int* nn2i = (int*)alloc((size_t)B * N * 3);
  float* nn2w = alloc((size_t)B * N * 3);
  float* fp2h1 = alloc((size_t)B * N * 256);
  float* fp2h2 = alloc((size_t)B * N * 128);
  float* headh = fp2h1;  // reuse (fp2h1 dead by head1)

  auto pad4 = [](int x) { return (x + 3) & ~3; };
  auto pad16 = [](int x) { return (x + 15) & ~15; };
  auto packW = [&](const float* W, int cin, int cout) -> const float* {
    int cpd = pad4(cin), co = pad16(cout);
    float* Wp = alloc((size_t)cpd * co);
    int tot = cpd * co;
    pack_weight_kernel<<<(tot + 255) / 256, 256, 0, stream>>>(W, Wp, cin,
                                                              cout, cpd, co);
    return Wp;
  };

  // pytree leaf order (dict keys sorted): fp2[1..8] fp3[9..16] head1[17..20]
  // head2[21..22] sa2[23..46] sa3[47..70]
  auto mkSA = [&](int base, int ci0, int c0, int c1, int c2) {
    SAParams p{};
    int cins[3] = {ci0, c0, c1};
    int couts[3] = {c0, c1, c2};
    for (int l = 0; l < 3; ++l) {
      p.Wp[l] = packW(F(base + 4 * l + 0), cins[l], couts[l]);
      p.bp[l] = F(base + 4 * l + 1);
      p.sp[l] = F(base + 4 * l + 2);
      p.zp[l] = F(base + 4 * l + 3);
      p.cin[l] = cins[l];
      p.cout[l] = couts[l];
      p.coP[l] = pad16(couts[l]);
    }
    return p;
  };
  SAParams pA = mkSA(23, 9, 64, 64, 128);
  SAParams pB = mkSA(35, 9, 64, 96, 128);
  SAParams pC = mkSA(47, 259, 128, 196, 256);
  SAParams pD = mkSA(59, 259, 128, 196, 256);
  const float* Wfp3a = packW(F(9), 768, 256);
  const float* Wfp3b = packW(F(13), 256, 256);
  const float* Wfp2a = packW(F(1), 262, 256);
  const float* Wfp2b = packW(F(5), 256, 128);
  const float* Whd1 = packW(F(17), 128, 128);

  size_t saSm = (size_t)2 * 32 * SASTRIDE * sizeof(float);
  auto mlpSm = [&](int cin) {
    return (size_t)16 * (pad4(cin) + 4) * sizeof(float);
  };

  // 1) transpose input
  {
    int tot = B * C * N;
    prep_kernel<<<(tot + 255) / 256, 256, 0, stream>>>(xyz, l1_xyz, l1_pts, B,
                                                       C, N);
  }
  // 2) SA level 2
  fps_kernel<<<B, 256, (size_t)(N + 512) * sizeof(float), stream>>>(
      l1_xyz, l2_xyz, N, 512);
  {
    int tot = B * 512;
    ball_query_kernel<<<(tot + 127) / 128, 128, 0, stream>>>(
        l1_xyz, l2_xyz, idx2A, B, N, 512, 16, 0.01f);
    ball_query_kernel<<<(tot + 127) / 128, 128, 0, stream>>>(
        l1_xyz, l2_xyz, idx2B, B, N, 512, 32, 0.04f);
  }
  sa_branch_kernel<<<B * 512, 128, saSm, stream>>>(
      l1_xyz, l1_pts, l2_xyz, idx2A, N, 512, 16, 6, pA, l2_pts, 0, 256);
  sa_branch_kernel<<<B * 512, 128, saSm, stream>>>(
      l1_xyz, l1_pts, l2_xyz, idx2B, N, 512, 32, 6, pB, l2_pts, 128, 256);
  // 3) SA level 3
  fps_kernel<<<B, 256, (size_t)(512 + 512) * sizeof(float), stream>>>(
      l2_xyz, l3_xyz, 512, 128);
  {
    int tot = B * 128;
    ball_query_kernel<<<(tot + 127) / 128, 128, 0, stream>>>(
        l2_xyz, l3_xyz, idx3A, B, 512, 128, 16, 0.04f);
    ball_query_kernel<<<(tot + 127) / 128, 128, 0, stream>>>(
        l2_xyz, l3_xyz, idx3B, B, 512, 128, 32, 0.16f);
  }
  sa_branch_kernel<<<B * 128, 128, saSm, stream>>>(
      l2_xyz, l2_pts, l3_xyz, idx3A, 512, 128, 16, 256, pC, l3_pts, 0, 512);
  sa_branch_kernel<<<B * 128, 128, saSm, stream>>>(
      l2_xyz, l2_pts, l3_xyz, idx3B, 512, 128, 32, 256, pD, l3_pts, 256, 512);
  // 4) fp3: fused 3-NN interp + MLP 768->256->256
  {
    int tot = B * 512;
    knn3_kernel<<<(tot + 127) / 128, 128, 0, stream>>>(l2_xyz, l3_xyz, B, 512,
                                                       128, nn3i, nn3w);
    dim3 g1((B * 512) / 16, (256 + 127) / 128);
    mlp_layer_kernel<<<g1, 128, mlpSm(768), stream>>>(
        nullptr, l2_pts, 256, l3_pts, 512, nn3i, nn3w, Wfp3a, F(10), F(11),
        F(12), fp3h1, B * 512, 768, 256, 256);
    mlp_layer_kernel<<<g1, 128, mlpSm(256), stream>>>(
        fp3h1, nullptr, 0, nullptr, 0, nullptr, nullptr, Wfp3b, F(14), F(15),
        F(16), fp3h2, B * 512, 256, 256, 256);
  }
  // 5) fp2: fused 3-NN interp + MLP 262->256->128, then head1
  {
    int tot = B * N;
    knn3_kernel<<<(tot + 127) / 128, 128, 0, stream>>>(l1_xyz, l2_xyz, B, N,
                                                       512, nn2i, nn2w);
    dim3 g1((B * N) / 16, (256 + 127) / 128);
    mlp_layer_kernel<<<g1, 128, mlpSm(262), stream>>>(
        nullptr, l1_pts, 6, fp3h2, 256, nn2i, nn2w, Wfp2a, F(2), F(3), F(4),
        fp2h1, B * N, 262, 256, 256);
    dim3 g2((B * N) / 16, 1);
    mlp_layer_kernel<<<g2, 128, mlpSm(256), stream>>>(
        fp2h1, nullptr, 0, nullptr, 0, nullptr, nullptr, Wfp2b, F(6), F(7),
        F(8), fp2h2, B * N, 256, 128, 128);
    mlp_layer_kernel<<<g2, 128, mlpSm(128), stream>>>(
        fp2h2, nullptr, 0, nullptr, 0, nullptr, nullptr, Whd1, F(18), F(19),
        F(20), headh, B * N, 128, 128, 128);
  }
  // 6) head2 sigmoid -> d_out[0 : B*N)
  head2_kernel<<<(B * N + 255) / 256, 256, 0, stream>>>(
      headh, F(21), F(22), (float*)d_out, B * N, 128);
  // 7) l3_points transpose -> d_out[B*N : )
  {
    int tot = B * 128 * 512;
    out2_kernel<<<(tot + 255) / 256, 256, 0, stream>>>(
        l3_pts, (float*)d_out + (size_t)B * N, B, 128, 512);
  }
}